// Embedding_41180146434078
// MI455X (gfx1250) — compile-verified
//
#include <hip/hip_runtime.h>

// diag-embed: out[i, j, c] = x[i, j] * (j == c)
// x: f32[8192, 176], out: f32[8192, 176, 176]  (~1.015 GB of stores, ~0 FLOPs)
//
// Pure store-bandwidth kernel. Bulk zeros are streamed with the CDNA5 async
// LDS->global path (global_store_async_from_lds_b128, ASYNCcnt) with a
// non-temporal hint; the one chunk per row holding the diagonal element is
// written with a regular NT b128 store.

typedef float float4v __attribute__((ext_vector_type(4)));

#define D_MODEL 176
#define MAT4    7744u   // (176*176)/4 float4 chunks per output matrix
#define ROW4    44u     // 176/4 float4 chunks per output row

__global__ __launch_bounds__(256) void diag_embed_kernel(
    const float* __restrict__ x, float4v* __restrict__ out4, unsigned total4) {

  // 512 B of zeros in LDS; lane L sources its async b128 stores from zbuf[L].
  __shared__ float4v zbuf[32];
  if (threadIdx.x < 32) zbuf[threadIdx.x] = (float4v){0.f, 0.f, 0.f, 0.f};
  __syncthreads();  // LDS writes committed before the async engine reads them

  const unsigned lane = threadIdx.x & 31u;
  // Generic pointers into LDS carry the LDS byte offset in their low 32 bits
  // (per the ISA's LDS aperture mapping).
  const unsigned lds_off =
      (unsigned)(unsigned long long)(uintptr_t)&zbuf[lane];

  const unsigned stride = gridDim.x * blockDim.x;
  for (unsigned k = blockIdx.x * blockDim.x + threadIdx.x; k < total4;
       k += stride) {
    unsigned i    = k / MAT4;          // batch index
    unsigned rem4 = k - i * MAT4;      // float4 index within 176x176 matrix
    unsigned j    = rem4 / ROW4;       // output row
    unsigned c4   = rem4 - j * ROW4;   // float4 column chunk (cols 4*c4..4*c4+3)

    float4v* dst = out4 + k;
    if ((j >> 2) == c4) {
      // This 16B chunk contains the diagonal element (i, j, j).
      float v = x[i * D_MODEL + j];
      unsigned l = j & 3u;
      float4v t;
      t.x = (l == 0u) ? v : 0.f;
      t.y = (l == 1u) ? v : 0.f;
      t.z = (l == 2u) ? v : 0.f;
      t.w = (l == 3u) ? v : 0.f;
      __builtin_nontemporal_store(t, dst);  // global_store_b128 th:TH_STORE_NT
    } else {
      // Pure-zero chunk: stream 16 B from the LDS zero block, async,
      // non-temporal (1 GB output >> 192 MB L2; never re-read).
      unsigned long long ga = (unsigned long long)(uintptr_t)dst;
      asm volatile(
          "global_store_async_from_lds_b128 %0, %1, off th:TH_STORE_NT"
          :
          : "v"(ga), "v"(lds_off)
          : "memory");
    }
  }

  // Drain outstanding async stores before wave termination.
  asm volatile("s_wait_asynccnt 0x0" ::: "memory");
}

extern "C" void kernel_launch(void* const* d_in, const int* in_sizes, int n_in,
                              void* d_out, int out_size, void* d_ws,
                              size_t ws_size, hipStream_t stream) {
  const float* x  = (const float*)d_in[0];
  float4v* out4   = (float4v*)d_out;
  unsigned total4 = (unsigned)(out_size / 4);  // 63,438,848 float4 chunks

  dim3 block(256);  // 8 wave32 waves per workgroup
  dim3 grid(8192);  // grid-stride: ~30 chunks/thread, plenty of stores in flight
  diag_embed_kernel<<<grid, block, 0, stream>>>(x, out4, total4);
}